// VectorizedEmbeddingCache_24412594111159
// MI455X (gfx1250) — compile-verified
//
#include <hip/hip_runtime.h>
#include <hip/hip_bf16.h>

// ---------------------------------------------------------------------------
// Embedding cache lookup with LRU miss handling.
// Heavy kernel uses CDNA5 async global<->LDS data-mover instructions
// (GLOBAL_LOAD_ASYNC_TO_LDS_B128 / GLOBAL_STORE_ASYNC_FROM_LDS_B128,
//  tracked by ASYNCcnt, waited with s_wait_asynccnt). Wave32 everywhere.
// Batched 4 rows/wave: 4 overlapped async gathers, one wait, 4 async stores.
// ---------------------------------------------------------------------------

#define SCAN_TPB  256
#define SCAN_EPT  8
#define SCAN_TILE (SCAN_TPB * SCAN_EPT)   // 2048 elements per block

#define ROWS_PER_WAVE 4
#define WAVES_PER_BLK 8
#define ROW_BYTES     512                 // D=128 floats

// ---- phase 0: init workspace ------------------------------------------------
__global__ void k_init(int* __restrict__ mark, int* __restrict__ slot_newid,
                       int V, int M) {
  int i = blockIdx.x * blockDim.x + threadIdx.x;
  if (i < V) mark[i] = 0;
  if (i < M) slot_newid[i] = -1;
}

// ---- phase 1: mark missed ids (idempotent stores; races benign) -------------
__global__ void k_mark(const int* __restrict__ flat,
                       const int* __restrict__ mapping,
                       int* __restrict__ mark, int N) {
  int i = blockIdx.x * blockDim.x + threadIdx.x;
  if (i < N) {
    int id = flat[i];
    if (mapping[id] < 0) mark[id] = 1;
  }
}

// ---- phase 2: evict_order = argsort(access_tick) = inverse permutation ------
__global__ void k_evict(const int* __restrict__ access_tick,
                        int* __restrict__ evict_order, int M) {
  int s = blockIdx.x * blockDim.x + threadIdx.x;
  if (s < M) {
    int t = access_tick[s];
    if (t >= 0 && t < M) evict_order[t] = s;
  }
}

// ---- phase 3a: per-block reduction of mark[] --------------------------------
__global__ __launch_bounds__(SCAN_TPB)
void k_reduce(const int* __restrict__ mark, int* __restrict__ blocksums, int V) {
  __shared__ int sh[SCAN_TPB];
  int base = blockIdx.x * SCAN_TILE + threadIdx.x * SCAN_EPT;
  int s = 0;
#pragma unroll
  for (int j = 0; j < SCAN_EPT; ++j) {
    int v = base + j;
    if (v < V) s += mark[v];
  }
  sh[threadIdx.x] = s;
  __syncthreads();
  for (int off = SCAN_TPB >> 1; off > 0; off >>= 1) {
    if (threadIdx.x < off) sh[threadIdx.x] += sh[threadIdx.x + off];
    __syncthreads();
  }
  if (threadIdx.x == 0) blocksums[blockIdx.x] = sh[0];
}

// ---- phase 3b: tiny serial exclusive scan of blocksums (nb ~ 489) -----------
__global__ void k_scan_bs(int* __restrict__ blocksums, int nb,
                          int* __restrict__ num_unique) {
  if (blockIdx.x == 0 && threadIdx.x == 0) {
    int acc = 0;
    for (int b = 0; b < nb; ++b) {
      int t = blocksums[b];
      blocksums[b] = acc;
      acc += t;
    }
    *num_unique = acc;
  }
}

// ---- phase 3c: recompute local scan, assign slots to unique missed ids ------
__global__ __launch_bounds__(SCAN_TPB)
void k_assign(const int* __restrict__ mark, const int* __restrict__ blocksums,
              const int* __restrict__ num_unique_p, const int* __restrict__ filled_p,
              const int* __restrict__ evict_order, int* __restrict__ slot_newid,
              int V, int M) {
  __shared__ int sh[SCAN_TPB];
  int base = blockIdx.x * SCAN_TILE + threadIdx.x * SCAN_EPT;
  int loc[SCAN_EPT];
  int s = 0;
#pragma unroll
  for (int j = 0; j < SCAN_EPT; ++j) {
    int v = base + j;
    int m = (v < V) ? mark[v] : 0;
    loc[j] = s;          // exclusive within this thread
    s += m;
  }
  sh[threadIdx.x] = s;
  __syncthreads();
  // Hillis-Steele inclusive scan across the block
  for (int off = 1; off < SCAN_TPB; off <<= 1) {
    int t = (threadIdx.x >= off) ? sh[threadIdx.x - off] : 0;
    __syncthreads();
    sh[threadIdx.x] += t;
    __syncthreads();
  }
  int texcl = (threadIdx.x == 0) ? 0 : sh[threadIdx.x - 1];
  int boff  = blocksums[blockIdx.x];

  int nuniq  = *num_unique_p;
  int filled = *filled_p;
  int free_cap = M - filled; if (free_cap < 0) free_cap = 0;
  int alloc_free = (nuniq < free_cap) ? nuniq : free_cap;

#pragma unroll
  for (int j = 0; j < SCAN_EPT; ++j) {
    int v = base + j;
    if (v < V && mark[v]) {
      int pos = boff + texcl + loc[j];           // rank among sorted unique misses
      int slot;
      if (pos < alloc_free) {
        slot = filled + pos;
        if (slot > M - 1) slot = M - 1;
      } else {
        int e = pos - alloc_free;
        if (e > M - 1) e = M - 1;
        slot = evict_order[e];                   // LRU victim
      }
      slot_newid[slot] = v;                      // slot now holds embedding v
    }
  }
}

// ---- phase 4: bandwidth kernel ----------------------------------------------
// One wave32 handles ROWS_PER_WAVE rows of 512B each:
//   issue 4 async gather loads (distinct LDS buffers)  -> latencies overlap
//   s_wait_asynccnt 0                                  -> only loads outstanding
//   issue 4 async stores (fire-and-forget; S_ENDPGM's implicit wait-idle drains)
__global__ __launch_bounds__(32 * WAVES_PER_BLK)
void k_out(const int* __restrict__ flat, const float* __restrict__ cpu_table,
           const float* __restrict__ cache, const int* __restrict__ mapping,
           const int* __restrict__ slot_newid, float* __restrict__ out, int N) {
  __shared__ char lds_stage[WAVES_PER_BLK * ROWS_PER_WAVE * ROW_BYTES]; // 16 KB
  const int wave = threadIdx.x >> 5;
  const int lane = threadIdx.x & 31;
  const int base = (blockIdx.x * WAVES_PER_BLK + wave) * ROWS_PER_WAVE;
  if (base >= N) return;                         // whole wave exits together

  // LDS byte offset = low 32 bits of the flat shared-aperture address
  const unsigned lds0 =
      (unsigned)(unsigned long long)(const void*)
          &lds_stage[wave * (ROWS_PER_WAVE * ROW_BYTES) + lane * 16];

#pragma unroll
  for (int r = 0; r < ROWS_PER_WAVE; ++r) {
    const int row = base + r;                    // uniform per wave
    if (row < N) {
      const int id = flat[row];
      const int s0 = mapping[id];
      const float* src;
      if (s0 < 0) {
        // miss: cache slot was just filled with cpu_table[id] -> read directly
        src = cpu_table + (size_t)id * 128;
      } else {
        int nid = slot_newid[s0];
        // hit whose slot was stolen by the miss-scatter reads the NEW row
        // (reference updates cache before the final gather)
        src = (nid >= 0) ? cpu_table + (size_t)nid * 128
                         : cache     + (size_t)s0  * 128;
      }
      unsigned long long gsrc = (unsigned long long)(const void*)src + lane * 16u;
      unsigned           ldst = lds0 + (unsigned)(r * ROW_BYTES);
      asm volatile("global_load_async_to_lds_b128 %0, %1, off"
                   :: "v"(ldst), "v"(gsrc) : "memory");
    }
  }

  // All outstanding ASYNC ops are loads -> cnt==0 means every buffer is filled.
  asm volatile("s_wait_asynccnt 0" ::: "memory");

#pragma unroll
  for (int r = 0; r < ROWS_PER_WAVE; ++r) {
    const int row = base + r;
    if (row < N) {
      unsigned long long gdst =
          (unsigned long long)(const void*)(out) + (size_t)row * ROW_BYTES + lane * 16u;
      unsigned ldsr = lds0 + (unsigned)(r * ROW_BYTES);
      asm volatile("global_store_async_from_lds_b128 %0, %1, off"
                   :: "v"(gdst), "v"(ldsr) : "memory");
    }
  }
  // S_ENDPGM performs an implicit wait-idle, draining the async stores before
  // the wave (and its LDS allocation) is released.
}

// ---------------------------------------------------------------------------
extern "C" void kernel_launch(void* const* d_in, const int* in_sizes, int n_in,
                              void* d_out, int out_size, void* d_ws, size_t ws_size,
                              hipStream_t stream) {
  // setup_inputs order: indices, cpu_table, cache_data, mapping_table,
  //                     access_tick, slot_to_id, filled
  const int*   indices     = (const int*)  d_in[0];
  const float* cpu_table   = (const float*)d_in[1];
  const float* cache_data  = (const float*)d_in[2];
  const int*   mapping     = (const int*)  d_in[3];
  const int*   access_tick = (const int*)  d_in[4];
  /* d_in[5] slot_to_id: only affects final mapping state, not the output */
  const int*   filled_p    = (const int*)  d_in[6];

  const int N = in_sizes[0];                    // B*L flat lookups
  const int D = (N > 0) ? (out_size / N) : 128; // embedding dim (128)
  const int V = in_sizes[1] / D;                // total embeddings
  const int M = in_sizes[2] / D;                // cache entries

  // carve workspace (ints): mark[V] | blocksums[nb] | num_unique | evict[M] | slot_newid[M]
  const int nb = (V + SCAN_TILE - 1) / SCAN_TILE;
  char* ws = (char*)d_ws;
  int* mark       = (int*)ws; ws += (size_t)V * 4;
  int* blocksums  = (int*)ws; ws += (size_t)((nb + 15) & ~15) * 4;
  int* num_unique = (int*)ws; ws += 64;
  int* evict      = (int*)ws; ws += (size_t)M * 4;
  int* slot_newid = (int*)ws; ws += (size_t)M * 4;

  const int TPB = 256;
  const int mx = (V > M) ? V : M;

  k_init   <<<(mx + TPB - 1) / TPB, TPB, 0, stream>>>(mark, slot_newid, V, M);
  k_mark   <<<(N  + TPB - 1) / TPB, TPB, 0, stream>>>(indices, mapping, mark, N);
  k_evict  <<<(M  + TPB - 1) / TPB, TPB, 0, stream>>>(access_tick, evict, M);
  k_reduce <<<nb, SCAN_TPB, 0, stream>>>(mark, blocksums, V);
  k_scan_bs<<<1, 64, 0, stream>>>(blocksums, nb, num_unique);
  k_assign <<<nb, SCAN_TPB, 0, stream>>>(mark, blocksums, num_unique, filled_p,
                                         evict, slot_newid, V, M);

  const int rows_per_blk = WAVES_PER_BLK * ROWS_PER_WAVE;  // 32 rows / block
  k_out    <<<(N + rows_per_blk - 1) / rows_per_blk, 32 * WAVES_PER_BLK, 0, stream>>>(
      indices, cpu_table, cache_data, mapping, slot_newid, (float*)d_out, N);
}